// DGCNN_8839042695335
// MI455X (gfx1250) — compile-verified
//
#include <hip/hip_runtime.h>
#include <math.h>

#define CDIV(a,b) (((a)+(b)-1)/(b))
#define FINF 3.402823466e38f

typedef __attribute__((ext_vector_type(2))) float v2f;
typedef __attribute__((ext_vector_type(8))) float v8f;

// ---------------------------------------------------------------------------
// avg pool along points:  t[b,tp,c] = mean_j y[b, tp*p+j, c]
// ---------------------------------------------------------------------------
__global__ void avgpool_kernel(const float* __restrict__ y, float* __restrict__ t,
                               int Nsrc, int Nt, int cin, int p, int total) {
  for (int o = blockIdx.x * blockDim.x + threadIdx.x; o < total; o += gridDim.x * blockDim.x) {
    int c = o % cin; int rest = o / cin; int tp = rest % Nt; int b = rest / Nt;
    float s = 0.f;
    for (int j = 0; j < p; ++j) s += y[((size_t)b * Nsrc + tp * p + j) * cin + c];
    t[o] = s / (float)p;
  }
}

// ---------------------------------------------------------------------------
// KNN: one workgroup per (b, target). Distances in LDS, k argmin passes.
// Tie-break: smaller source index (matches top_k stability).
// ---------------------------------------------------------------------------
template<int CIN>
__global__ void knn_kernel(const float* __restrict__ y, const float* __restrict__ t,
                           int* __restrict__ idxout, int Nsrc, int Nt, int k) {
  extern __shared__ float sm[];
  float* dist = sm;                    // Nsrc
  float* redv = dist + Nsrc;           // blockDim
  int*   redi = (int*)(redv + blockDim.x);
  __shared__ float tvec[CIN];
  int tid = threadIdx.x;
  int b = blockIdx.x / Nt, tt = blockIdx.x % Nt;
  if (tid < CIN) tvec[tid] = t[((size_t)b * Nt + tt) * CIN + tid];
  __syncthreads();
  for (int j = tid; j < Nsrc; j += blockDim.x) {
    const float* yp = y + ((size_t)b * Nsrc + j) * CIN;
    float d = 0.f;
#pragma unroll
    for (int c = 0; c < CIN; ++c) { float df = tvec[c] - yp[c]; d += df * df; }
    dist[j] = d;
  }
  __syncthreads();
  for (int kk = 0; kk < k; ++kk) {
    float bv = FINF; int bi = 0x7fffffff;
    for (int j = tid; j < Nsrc; j += blockDim.x)
      if (dist[j] < bv) { bv = dist[j]; bi = j; }
    redv[tid] = bv; redi[tid] = bi;
    __syncthreads();
    for (int s = blockDim.x >> 1; s > 0; s >>= 1) {
      if (tid < s) {
        float v2 = redv[tid + s]; int i2 = redi[tid + s];
        if (v2 < redv[tid] || (v2 == redv[tid] && i2 < redi[tid])) { redv[tid] = v2; redi[tid] = i2; }
      }
      __syncthreads();
    }
    if (tid == 0) { int w = redi[0]; idxout[((size_t)b * Nt + tt) * k + kk] = w; dist[w] = FINF; }
    __syncthreads();
  }
}

// ---------------------------------------------------------------------------
// Prepack stacked stage weights, PAIR-INTERLEAVED for b64 fragment loads:
//   w2p[(r/2)*U*2 + c*2 + (r&1)] = r<cin ? w_rel[r][c] : r<2cin ? w_abs[r-cin][c] : 0
// and bias2 = b_rel + b_abs.
// ---------------------------------------------------------------------------
__global__ void pack_w2_kernel(const float* __restrict__ wrel, const float* __restrict__ wabs,
                               const float* __restrict__ brel, const float* __restrict__ babs,
                               float* __restrict__ w2p, float* __restrict__ bias2,
                               int cin, int u, int st) {
  int total = st * u;
  for (int o = blockIdx.x * blockDim.x + threadIdx.x; o < total; o += gridDim.x * blockDim.x) {
    int r = o / u, c = o - r * u;
    float v = 0.f;
    if (r < cin) v = wrel[r * u + c];
    else if (r < 2 * cin) v = wabs[(r - cin) * u + c];
    w2p[(r >> 1) * u * 2 + c * 2 + (r & 1)] = v;
    if (o < u) bias2[o] = brel[o] + babs[o];
  }
}

// generic pair-interleave pack:  Wp[(r/2)*N*2 + c*2 + (r&1)] = W[r][c]
__global__ void pack_pairs_kernel(const float* __restrict__ W, float* __restrict__ Wp,
                                  int K, int N) {
  int total = K * N;
  for (int o = blockIdx.x * blockDim.x + threadIdx.x; o < total; o += gridDim.x * blockDim.x) {
    int r = o / N, c = o - r * N;
    Wp[(r >> 1) * N * 2 + c * 2 + (r & 1)] = W[o];
  }
}

// ---------------------------------------------------------------------------
// Fused stage kernel (templated -> fully unrolled, branch-free).
// One workgroup per (b, target).  A in LDS = [y[idx]-t || t] (zero padded),
// B = pair-packed w2.  z = A@B + bias2 -> per-channel sum/sumsq + max over k.
// Each wave owns one M-strip and TPW N-tile accumulators; A fragment loaded
// once per K-step (single b64) and reused.  Epilogue writes per-(mtile,half)
// partials to UNIQUE LDS slots (no atomics, no divergent branches), combined
// deterministically after the barrier.
// ---------------------------------------------------------------------------
template<int CIN, int U, int KNB, int KPAD>
__global__ __launch_bounds__(256) void stage_gemm_kernel(
    const float* __restrict__ ysrc, const float* __restrict__ tb,
    const int* __restrict__ idx, const float* __restrict__ w2p,
    const float* __restrict__ bias2,
    float* __restrict__ zmax, float* __restrict__ gsum, float* __restrict__ gsq,
    int Nsrc, int Nt) {
  constexpr int ST = (2 * CIN + 3) & ~3;
  constexpr int NTM = KPAD / 16;
  constexpr int NTN = U / 16;
  constexpr int TILES = NTM * NTN;
  constexpr int TPW = (TILES + 7) / 8;   // tiles per wave (divides NTN for all stages)
  constexpr int SLOTS = NTM * 2;
  __shared__ float patch[KPAD * ST];
  __shared__ float psum[SLOTS][U], psq[SLOTS][U], pmax[SLOTS][U];
  __shared__ int sidx[KNB];
  int tid = threadIdx.x;
  int b = blockIdx.x / Nt, tt = blockIdx.x % Nt;
  const float* myt = tb + ((size_t)b * Nt + tt) * CIN;

  if (tid < KNB) sidx[tid] = idx[((size_t)b * Nt + tt) * KNB + tid];
  __syncthreads();

  for (int e = tid; e < KPAD * ST; e += 256) {
    int kk = e / ST, c = e - kk * ST;
    float v = 0.f;
    if (kk < KNB && c < 2 * CIN) {
      if (c < CIN) v = ysrc[((size_t)b * Nsrc + sidx[kk]) * CIN + c] - myt[c];
      else         v = myt[c - CIN];
    }
    patch[e] = v;
  }
  __syncthreads();

  int l = tid & 31;
  int wave = tid >> 5;
  int arow = l & 15, koff = (l >> 4) << 1, colL = l & 15, half = l >> 4;
  int t0 = wave * TPW;
  if (t0 < TILES) {
    int m0 = (t0 / NTN) << 4;
    int nbase = (t0 % NTN) << 4;
    v8f acc[TPW];
#pragma unroll
    for (int j = 0; j < TPW; ++j) acc[j] = (v8f){};
#pragma unroll
    for (int kc = 0; kc < ST; kc += 4) {
      int kb = kc + koff;
      v2f a = *(const v2f*)(&patch[(m0 + arow) * ST + kb]);   // ds b64 (aligned)
      int prow = kb >> 1;
#pragma unroll
      for (int j = 0; j < TPW; ++j) {
        int col = nbase + (j << 4) + colL;
        v2f bb = *(const v2f*)(w2p + ((size_t)prow * U + col) * 2);  // global b64
        acc[j] = __builtin_amdgcn_wmma_f32_16x16x4_f32(false, a, false, bb, (short)0, acc[j], false, false);
      }
    }
    int mbase = m0 + (half << 3);
    int slot = ((m0 >> 4) << 1) + half;   // unique (mtile, half) slot
#pragma unroll
    for (int j = 0; j < TPW; ++j) {
      int col = nbase + (j << 4) + colL;
      float bias = bias2[col];
      float lsum = 0.f, lsq = 0.f, lmax = -FINF;
#pragma unroll
      for (int r = 0; r < 8; ++r) {
        int m = mbase + r;
        float v = acc[j][r] + bias;
        bool ok = (m < KNB);               // compile-time true for stages 1-3
        float vv = ok ? v : 0.f;
        lsum += vv;
        lsq += vv * vv;
        lmax = fmaxf(lmax, ok ? v : -FINF);
      }
      psum[slot][col] = lsum;
      psq[slot][col] = lsq;
      pmax[slot][col] = lmax;
    }
  }
  __syncthreads();
  for (int c = tid; c < U; c += 256) {
    float s = 0.f, q = 0.f, m = -FINF;
#pragma unroll
    for (int s2 = 0; s2 < SLOTS; ++s2) {
      s += psum[s2][c];
      q += psq[s2][c];
      m = fmaxf(m, pmax[s2][c]);
    }
    zmax[((size_t)b * Nt + tt) * U + c] = m;
    atomicAdd(&gsum[c], s);
    atomicAdd(&gsq[c], q);
  }
}

// coalesced column partial stats: threads = consecutive channels, loop row chunk
__global__ void bn_col_partial_kernel(const float* __restrict__ z, float* __restrict__ gsum,
                                      float* __restrict__ gsq, int M, int N, int rpb) {
  int ch = blockIdx.x * blockDim.x + threadIdx.x;
  int r0 = blockIdx.y * rpb;
  int r1 = min(r0 + rpb, M);
  float s = 0.f, q = 0.f;
  for (int r = r0; r < r1; ++r) { float v = z[(size_t)r * N + ch]; s += v; q += v * v; }
  atomicAdd(&gsum[ch], s);
  atomicAdd(&gsq[ch], q);
}

// BN finalize from atomic-accumulated sums
__global__ void bn_finalize_kernel(const float* __restrict__ sum, const float* __restrict__ sq,
    const float* __restrict__ g, const float* __restrict__ be,
    float* __restrict__ scale, float* __restrict__ shift, int u, float cnt) {
  int c = blockIdx.x * blockDim.x + threadIdx.x;
  if (c < u) {
    float mean = sum[c] / cnt;
    float var  = sq[c] / cnt - mean * mean;
    float sc = g[c] * rsqrtf(var + 1e-3f);
    scale[c] = sc; shift[c] = be[c] - mean * sc;
  }
}

__global__ void apply_bnrelu_inplace_kernel(float* __restrict__ y, const float* __restrict__ scale,
                                            const float* __restrict__ shift, int total, int u) {
  for (int o = blockIdx.x * blockDim.x + threadIdx.x; o < total; o += gridDim.x * blockDim.x) {
    int c = o % u;
    y[o] = fmaxf(y[o] * scale[c] + shift[c], 0.f);
  }
}

__global__ void pool_concat_kernel(const float* __restrict__ y, float* __restrict__ concat,
                                   int Nt, int u, int p2, int coff, int total) {
  for (int o = blockIdx.x * blockDim.x + threadIdx.x; o < total; o += gridDim.x * blockDim.x) {
    int c = o % u; int rest = o / u; int tp = rest % 256; int b = rest / 256;
    float m = -FINF;
    for (int j = 0; j < p2; ++j)
      m = fmaxf(m, y[((size_t)b * Nt + tp * p2 + j) * u + c]);
    concat[((size_t)b * 256 + tp) * 512 + coff + c] = m;
  }
}

// ---------------------------------------------------------------------------
// Generic f32 WMMA GEMM with pair-packed B:  C[M,N] = A[M,K] @ W[K,N] + bias.
// One wave per block; each wave owns TPN N-tiles: A fragment (b64) loaded once
// per K-step and reused across TPN packed-B (b64) WMMAs.
// ---------------------------------------------------------------------------
template<int TPN>
__global__ __launch_bounds__(32) void gemm_wmma_packed_kernel(
    const float* __restrict__ A, const float* __restrict__ Wp,
    const float* __restrict__ bias, float* __restrict__ C,
    int M, int K, int N) {
  int m0 = blockIdx.x << 4;
  int nbase = blockIdx.y * (TPN << 4);
  int l = threadIdx.x;
  int arow = l & 15, koff = (l >> 4) << 1, colL = l & 15, half = l >> 4;
  v8f acc[TPN];
#pragma unroll
  for (int j = 0; j < TPN; ++j) acc[j] = (v8f){};
  const float* arowp = A + (size_t)(m0 + arow) * K;
#pragma unroll 4
  for (int kc = 0; kc < K; kc += 4) {
    int kb = kc + koff;
    v2f a = *(const v2f*)(arowp + kb);                 // b64
    int prow = kb >> 1;
#pragma unroll
    for (int j = 0; j < TPN; ++j) {
      int col = nbase + (j << 4) + colL;
      v2f bb = *(const v2f*)(Wp + ((size_t)prow * N + col) * 2);  // b64
      acc[j] = __builtin_amdgcn_wmma_f32_16x16x4_f32(false, a, false, bb, (short)0, acc[j], false, false);
    }
  }
  int mbase = m0 + (half << 3);
#pragma unroll
  for (int j = 0; j < TPN; ++j) {
    int col = nbase + (j << 4) + colL;
    float bv = bias[col];
#pragma unroll
    for (int r = 0; r < 8; ++r)
      C[(size_t)(mbase + r) * N + col] = acc[j][r] + bv;
  }
}

// y3[b,ch] = max_pt relu(bn(z2[b,pt,ch]))  ( = relu(max_pt bn) exactly )
__global__ void rowmax_relu_kernel(const float* __restrict__ z, const float* __restrict__ scale,
    const float* __restrict__ shift, float* __restrict__ y3, int P, int N, int total) {
  for (int o = blockIdx.x * blockDim.x + threadIdx.x; o < total; o += gridDim.x * blockDim.x) {
    int ch = o % N; int b = o / N;
    float sc = scale[ch], sh = shift[ch];
    float m = -FINF;
    for (int pt = 0; pt < P; ++pt)
      m = fmaxf(m, z[((size_t)b * P + pt) * N + ch] * sc + sh);
    y3[o] = fmaxf(m, 0.f);
  }
}

__global__ void bnrelu_kernel(const float* __restrict__ z, const float* __restrict__ scale,
    const float* __restrict__ shift, float* __restrict__ y, int total, int N) {
  for (int o = blockIdx.x * blockDim.x + threadIdx.x; o < total; o += gridDim.x * blockDim.x) {
    int c = o % N;
    y[o] = fmaxf(z[o] * scale[c] + shift[c], 0.f);
  }
}

// final 256->40 dense + softmax, one block per batch row
__global__ void out_softmax_kernel(const float* __restrict__ y, const float* __restrict__ w,
                                   const float* __restrict__ bias, float* __restrict__ out,
                                   int K, int N) {
  __shared__ float lg[64];
  int b = blockIdx.x, tid = threadIdx.x;
  for (int c = tid; c < N; c += blockDim.x) {
    float s = bias[c];
    for (int j = 0; j < K; ++j) s += y[(size_t)b * K + j] * w[(size_t)j * N + c];
    lg[c] = s;
  }
  __syncthreads();
  if (tid == 0) {
    float m = -FINF;
    for (int c = 0; c < N; ++c) m = fmaxf(m, lg[c]);
    float ssum = 0.f;
    for (int c = 0; c < N; ++c) { float e = expf(lg[c] - m); lg[c] = e; ssum += e; }
    for (int c = 0; c < N; ++c) out[(size_t)b * N + c] = lg[c] / ssum;
  }
}

// ---------------------------------------------------------------------------
extern "C" void kernel_launch(void* const* d_in, const int* in_sizes, int n_in,
                              void* d_out, int out_size, void* d_ws, size_t ws_size,
                              hipStream_t stream) {
  (void)in_sizes; (void)n_in; (void)out_size; (void)ws_size;
  const float* x = (const float*)d_in[0];
  float* out = (float*)d_out;
  float* ws = (float*)d_ws;
  const int B = 64;
  const int NP[5] = {1024, 512, 256, 256, 256};
  const int PK[4] = {40, 32, 32, 32};
  const int UU[4] = {64, 64, 128, 256};
  const int CI[4] = {3, 64, 64, 128};

  // workspace layout (float offsets)
  float* BUFY0  = ws;                     // 4,194,304
  float* BUFY1  = ws + 4194304;           // 4,194,304
  float* TB     = ws + 8388608;           // 2,097,152
  int*   IDXB   = (int*)(ws + 10485760);  // 1,310,720 ints
  float* SUM    = ws + 11796480;          // 1024
  float* SQ     = SUM + 1024;             // 1024
  float* SCALE  = SQ + 1024;              // 1024
  float* SHIFT  = SCALE + 1024;           // 1024
  float* CONCAT = ws + 11800576;          // 8,388,608  [64,256,512]
  float* Z2     = ws + 20189184;          // 16,777,216 [16384,1024]
  float* Y3     = ws + 36966400;          // 65,536
  float* Z4     = Y3 + 65536;             // 32,768
  float* Y4     = Z4 + 32768;             // 32,768
  float* Z5     = Y4 + 32768;             // 16,384
  float* Y5     = Z5 + 16384;             // 16,384
  float* W2     = ws + 37130240;          // 65,536 (max ST*U = 256*256, pair-packed)
  float* BIAS2  = W2 + 65536;             // 1,024
  float* WPACK  = BIAS2 + 1024;           // 524,288 (max K*N of dense layers)

  const float* ysrc = x;
  float* bufs[2] = {BUFY0, BUFY1};
  int cur = 0, coff = 0;
  for (int i = 0; i < 4; ++i) {
    int Nsrc = NP[i], Nt = NP[i + 1], k = PK[i], u = UU[i], cin = CI[i];
    int p = Nsrc / Nt;
    const float* wrel = (const float*)d_in[1 + i * 6 + 0];
    const float* brel = (const float*)d_in[1 + i * 6 + 1];
    const float* wabs = (const float*)d_in[1 + i * 6 + 2];
    const float* babs = (const float*)d_in[1 + i * 6 + 3];
    const float* g    = (const float*)d_in[1 + i * 6 + 4];
    const float* be   = (const float*)d_in[1 + i * 6 + 5];

    int tot_t = B * Nt * cin;
    avgpool_kernel<<<CDIV(tot_t, 256), 256, 0, stream>>>(ysrc, TB, Nsrc, Nt, cin, p, tot_t);

    size_t knnlds = (size_t)(Nsrc + 256) * sizeof(float) + 256 * sizeof(int);
    if (cin == 3)
      knn_kernel<3><<<B * Nt, 256, knnlds, stream>>>(ysrc, TB, IDXB, Nsrc, Nt, k);
    else if (cin == 64)
      knn_kernel<64><<<B * Nt, 256, knnlds, stream>>>(ysrc, TB, IDXB, Nsrc, Nt, k);
    else
      knn_kernel<128><<<B * Nt, 256, knnlds, stream>>>(ysrc, TB, IDXB, Nsrc, Nt, k);

    int st = (2 * cin + 3) & ~3;
    pack_w2_kernel<<<CDIV(st * u, 256), 256, 0, stream>>>(wrel, wabs, brel, babs, W2, BIAS2, cin, u, st);

    hipMemsetAsync(SUM, 0, 2048 * sizeof(float), stream);  // SUM + SQ contiguous

    switch (i) {
      case 0: stage_gemm_kernel<3, 64, 40, 48><<<B * Nt, 256, 0, stream>>>(
                  ysrc, TB, IDXB, W2, BIAS2, bufs[cur], SUM, SQ, Nsrc, Nt); break;
      case 1: stage_gemm_kernel<64, 64, 32, 32><<<B * Nt, 256, 0, stream>>>(
                  ysrc, TB, IDXB, W2, BIAS2, bufs[cur], SUM, SQ, Nsrc, Nt); break;
      case 2: stage_gemm_kernel<64, 128, 32, 32><<<B * Nt, 256, 0, stream>>>(
                  ysrc, TB, IDXB, W2, BIAS2, bufs[cur], SUM, SQ, Nsrc, Nt); break;
      default: stage_gemm_kernel<128, 256, 32, 32><<<B * Nt, 256, 0, stream>>>(
                  ysrc, TB, IDXB, W2, BIAS2, bufs[cur], SUM, SQ, Nsrc, Nt); break;
    }

    float cnt = (float)B * (float)Nt * (float)k;
    bn_finalize_kernel<<<CDIV(u, 256), 256, 0, stream>>>(SUM, SQ, g, be, SCALE, SHIFT, u, cnt);

    int toty = B * Nt * u;
    apply_bnrelu_inplace_kernel<<<CDIV(toty, 256), 256, 0, stream>>>(bufs[cur], SCALE, SHIFT, toty, u);

    int p2 = Nt / 256;
    int totc = B * 256 * u;
    pool_concat_kernel<<<CDIV(totc, 256), 256, 0, stream>>>(bufs[cur], CONCAT, Nt, u, p2, coff, totc);

    coff += u;
    ysrc = bufs[cur];
    cur ^= 1;
  }

  // point MLP 512 -> 1024 over 16384 rows
  const float* w_mlp = (const float*)d_in[25]; const float* b_mlp  = (const float*)d_in[26];
  const float* g_mlp = (const float*)d_in[27]; const float* be_mlp = (const float*)d_in[28];
  pack_pairs_kernel<<<CDIV(512 * 1024, 256), 256, 0, stream>>>(w_mlp, WPACK, 512, 1024);
  gemm_wmma_packed_kernel<4><<<dim3(16384 / 16, 1024 / 64), 32, 0, stream>>>(CONCAT, WPACK, b_mlp, Z2, 16384, 512, 1024);
  hipMemsetAsync(SUM, 0, 2048 * sizeof(float), stream);
  bn_col_partial_kernel<<<dim3(1024 / 256, 64), 256, 0, stream>>>(Z2, SUM, SQ, 16384, 1024, 256);
  bn_finalize_kernel<<<4, 256, 0, stream>>>(SUM, SQ, g_mlp, be_mlp, SCALE, SHIFT, 1024, 16384.f);
  rowmax_relu_kernel<<<CDIV(64 * 1024, 256), 256, 0, stream>>>(Z2, SCALE, SHIFT, Y3, 256, 1024, 64 * 1024);

  // FC1 1024 -> 512
  const float* w_fc1 = (const float*)d_in[29]; const float* b_fc1  = (const float*)d_in[30];
  const float* g_fc1 = (const float*)d_in[31]; const float* be_fc1 = (const float*)d_in[32];
  pack_pairs_kernel<<<CDIV(1024 * 512, 256), 256, 0, stream>>>(w_fc1, WPACK, 1024, 512);
  gemm_wmma_packed_kernel<4><<<dim3(4, 512 / 64), 32, 0, stream>>>(Y3, WPACK, b_fc1, Z4, 64, 1024, 512);
  hipMemsetAsync(SUM, 0, 2048 * sizeof(float), stream);
  bn_col_partial_kernel<<<dim3(512 / 256, 1), 256, 0, stream>>>(Z4, SUM, SQ, 64, 512, 64);
  bn_finalize_kernel<<<2, 256, 0, stream>>>(SUM, SQ, g_fc1, be_fc1, SCALE, SHIFT, 512, 64.f);
  bnrelu_kernel<<<CDIV(64 * 512, 256), 256, 0, stream>>>(Z4, SCALE, SHIFT, Y4, 64 * 512, 512);

  // FC2 512 -> 256
  const float* w_fc2 = (const float*)d_in[33]; const float* b_fc2  = (const float*)d_in[34];
  const float* g_fc2 = (const float*)d_in[35]; const float* be_fc2 = (const float*)d_in[36];
  pack_pairs_kernel<<<CDIV(512 * 256, 256), 256, 0, stream>>>(w_fc2, WPACK, 512, 256);
  gemm_wmma_packed_kernel<4><<<dim3(4, 256 / 64), 32, 0, stream>>>(Y4, WPACK, b_fc2, Z5, 64, 512, 256);
  hipMemsetAsync(SUM, 0, 2048 * sizeof(float), stream);
  bn_col_partial_kernel<<<dim3(256 / 256, 1), 256, 0, stream>>>(Z5, SUM, SQ, 64, 256, 64);
  bn_finalize_kernel<<<1, 256, 0, stream>>>(SUM, SQ, g_fc2, be_fc2, SCALE, SHIFT, 256, 64.f);
  bnrelu_kernel<<<CDIV(64 * 256, 256), 256, 0, stream>>>(Z5, SCALE, SHIFT, Y5, 64 * 256, 256);

  // output dense 256 -> 40 + softmax
  const float* w_out = (const float*)d_in[37]; const float* b_out = (const float*)d_in[38];
  out_softmax_kernel<<<64, 64, 0, stream>>>(Y5, w_out, b_out, out, 256, 40);
}